// mHCResidual_28965259444506
// MI455X (gfx1250) — compile-verified
//
#include <hip/hip_runtime.h>

typedef float v2f __attribute__((ext_vector_type(2)));
typedef float v8f __attribute__((ext_vector_type(8)));

#define BT   8192          // B*T = 4*2048
#define SS   4             // streams
#define DD   1024          // feature dim
#define BT_PER_BLOCK 4     // (b,t) rows per block -> bias regs reused 4x
#define WS_A    0          // 64 floats: per-lane A-matrix image (16x4, rows>=4 zero)
#define WS_PS   64         // 4 floats: alpha_pos * H_pos[r]
#define WS_BIAS 128        // 4096 floats: bias_res + bias_pos

// ---------------------------------------------------------------------------
// Prep: sinkhorn on 4x4 (1 thread), fold alpha_res & identity into A-image,
// combine biases (all threads).
// ---------------------------------------------------------------------------
__global__ __launch_bounds__(256) void sinkhorn_prep_kernel(
    const float* __restrict__ H_res, const float* __restrict__ H_pos,
    const float* __restrict__ alpha_res, const float* __restrict__ alpha_pos,
    const float* __restrict__ bias_res, const float* __restrict__ bias_pos,
    float* __restrict__ ws)
{
    int gid = blockIdx.x * blockDim.x + threadIdx.x;
    if (gid < SS * DD)
        ws[WS_BIAS + gid] = bias_res[gid] + bias_pos[gid];

    if (blockIdx.x == 0 && threadIdx.x == 0) {
        const float eps = 1e-8f;
        float P[4][4];
        for (int r = 0; r < 4; ++r) {
            float m = H_res[r * 4 + 0];
            for (int c = 1; c < 4; ++c) m = fmaxf(m, H_res[r * 4 + c]);
            for (int c = 0; c < 4; ++c) P[r][c] = __expf(H_res[r * 4 + c] - m);
        }
        for (int it = 0; it < 20; ++it) {
            for (int r = 0; r < 4; ++r) {                 // row normalize (axis=-1)
                float s = P[r][0] + P[r][1] + P[r][2] + P[r][3];
                float inv = 1.0f / (s + eps);
                for (int c = 0; c < 4; ++c) P[r][c] *= inv;
            }
            for (int c = 0; c < 4; ++c) {                 // col normalize (axis=-2)
                float s = P[0][c] + P[1][c] + P[2][c] + P[3][c];
                float inv = 1.0f / (s + eps);
                for (int r = 0; r < 4; ++r) P[r][c] *= inv;
            }
        }
        float ar = alpha_res[0];
        float ap = alpha_pos[0];
        // M[r][s] = alpha_res * H_proj[s][r] + (r==s)   (residual folded in)
        float M[4][4];
        for (int r = 0; r < 4; ++r)
            for (int s = 0; s < 4; ++s)
                M[r][s] = ar * P[s][r] + ((r == s) ? 1.0f : 0.0f);
        // A-matrix per-lane image for V_WMMA_F32_16X16X4_F32:
        // lane m (0..15): A[m][0], A[m][1] ; lane 16+m: A[m][2], A[m][3]
        for (int lane = 0; lane < 32; ++lane) {
            int h = lane >> 4, m = lane & 15;
            for (int j = 0; j < 2; ++j)
                ws[WS_A + lane * 2 + j] = (m < 4) ? M[m][2 * h + j] : 0.0f;
        }
        for (int r = 0; r < 4; ++r)
            ws[WS_PS + r] = ap * H_pos[r];
    }
}

// ---------------------------------------------------------------------------
// Main: one block per 4 (b,t) rows; wave w owns d-tiles [w*8, w*8+8) for all
// 4 rows. Per tile: D(16x16) = A(16x4 = folded mix, rows>=4 zero) x
//                   B(4x16 = x tile) + C(bias + alpha_pos*H_pos*layer seed)
//
// - Bias for the wave's 8 tiles (32 floats/lane) is preloaded into registers
//   and reused across the 4 (b,t) rows.
// - Accumulator tuple persists across tiles: A rows 4..7 are zero, so D rows
//   4..7 == C rows 4..7 (stay zero) -> no per-tile re-zeroing/mov chains.
// - Lanes 16..31 of acc[0..3] feed only rows M=8..11 (zero A rows, never
//   stored) -> seeds are computed unconditionally on all lanes, no exec masks
//   before the WMMA (EXEC all-ones requirement satisfied).
// ---------------------------------------------------------------------------
__global__ __launch_bounds__(256) void mhc_mix_wmma_kernel(
    const float* __restrict__ x,        // (BT, S, D)
    const float* __restrict__ lout,     // (BT, D)
    const float* __restrict__ ws,       // workspace (A image, posScale, bias)
    float* __restrict__ out)            // (BT, S, D)
{
    const int bt0  = blockIdx.x * BT_PER_BLOCK;
    const int wave = threadIdx.x >> 5;
    const int lane = threadIdx.x & 31;
    const int ln   = lane & 15;
    const int h    = lane >> 4;

    // A operand: folded mix matrix, identical for all waves (L2-resident)
    v2f a;
    a.x = ws[WS_A + lane * 2 + 0];
    a.y = ws[WS_A + lane * 2 + 1];

    const float ps0 = ws[WS_PS + 0];
    const float ps1 = ws[WS_PS + 1];
    const float ps2 = ws[WS_PS + 2];
    const float ps3 = ws[WS_PS + 3];
    const float* bias = ws + WS_BIAS;

    // Preload this wave's bias values: 8 tiles x 4 rows, reused for 4 bt rows.
    float breg[8][4];
    #pragma unroll
    for (int i = 0; i < 8; ++i) {
        const int dcol = (wave * 8 + i) * 16 + ln;
        #pragma unroll
        for (int r = 0; r < 4; ++r)
            breg[i][r] = bias[r * DD + dcol];
    }

    // Persistent accumulator; elements 4..7 stay zero across all WMMAs.
    v8f acc;
    acc[0] = 0.0f; acc[1] = 0.0f; acc[2] = 0.0f; acc[3] = 0.0f;
    acc[4] = 0.0f; acc[5] = 0.0f; acc[6] = 0.0f; acc[7] = 0.0f;

    #pragma unroll
    for (int j = 0; j < BT_PER_BLOCK; ++j) {
        const int bt = bt0 + j;
        const float* xb  = x    + (size_t)bt * (SS * DD);
        const float* lob = lout + (size_t)bt * DD;
        float*       ob  = out  + (size_t)bt * (SS * DD);

        #pragma unroll
        for (int i = 0; i < 8; ++i) {
            const int dcol = (wave * 8 + i) * 16 + ln;

            // B operand: lane half h supplies rows K=2h, 2h+1 -> x[s][dcol].
            // x / lout are streamed once -> non-temporal loads.
            v2f b;
            b.x = __builtin_nontemporal_load(&xb[(2 * h) * DD + dcol]);
            b.y = __builtin_nontemporal_load(&xb[(2 * h) * DD + DD + dcol]);

            const float l = __builtin_nontemporal_load(&lob[dcol]);

            // Seed rows 0..3 (fmacs write directly into the acc tuple).
            acc[0] = fmaf(ps0, l, breg[i][0]);
            acc[1] = fmaf(ps1, l, breg[i][1]);
            acc[2] = fmaf(ps2, l, breg[i][2]);
            acc[3] = fmaf(ps3, l, breg[i][3]);

            // D = A x B + C   (EXEC all-ones here)
            acc = __builtin_amdgcn_wmma_f32_16x16x4_f32(
                    false, a, false, b, (short)0, acc, false, false);

            // Rows r=0..3 live in lanes 0..15 of acc[0..3]; each store is a
            // contiguous 64B segment of out[bt, r, d0:d0+16].
            if (h == 0) {
                __builtin_nontemporal_store(acc[0], &ob[0 * DD + dcol]);
                __builtin_nontemporal_store(acc[1], &ob[1 * DD + dcol]);
                __builtin_nontemporal_store(acc[2], &ob[2 * DD + dcol]);
                __builtin_nontemporal_store(acc[3], &ob[3 * DD + dcol]);
            }
        }
    }
}

extern "C" void kernel_launch(void* const* d_in, const int* in_sizes, int n_in,
                              void* d_out, int out_size, void* d_ws, size_t ws_size,
                              hipStream_t stream) {
    const float* x     = (const float*)d_in[0]; // (B,T,S,D)
    const float* lout  = (const float*)d_in[1]; // (B,T,D)
    const float* H_res = (const float*)d_in[2]; // (S,S)
    const float* H_pos = (const float*)d_in[3]; // (S,1)
    const float* a_res = (const float*)d_in[4]; // scalar
    const float* a_pos = (const float*)d_in[5]; // scalar
    const float* b_res = (const float*)d_in[6]; // (S,D)
    const float* b_pos = (const float*)d_in[7]; // (S,D)
    float* out = (float*)d_out;
    float* ws  = (float*)d_ws;

    sinkhorn_prep_kernel<<<16, 256, 0, stream>>>(H_res, H_pos, a_res, a_pos,
                                                 b_res, b_pos, ws);
    mhc_mix_wmma_kernel<<<BT / BT_PER_BLOCK, 256, 0, stream>>>(x, lout, ws, out);
}